// QFeatureMapQuantumEnhancedFF_41729902248269
// MI455X (gfx1250) — compile-verified
//
#include <hip/hip_runtime.h>
#include <hip/hip_bf16.h>
#include <cstddef>

typedef float v2f __attribute__((ext_vector_type(2)));
typedef float v8f __attribute__((ext_vector_type(8)));

// ---------------------------------------------------------------------------
// Kernel 1: theta[j] = sum_k x[k] * W[k][j]   (x = inputs row 0)
// Implemented as M=16 WMMA GEMM with A rows replicated = x, so every row of
// the 16x16 f32 accumulator equals the 16-wide theta tile.
// Partition: blockIdx.x = column tile (16 cols), wave-in-block = K chunk (512).
// W (4096 x 4095 f32, 67 MB) is read exactly once, coalesced.
// ---------------------------------------------------------------------------
__global__ __launch_bounds__(256) void qeff_gemv_wmma(
    const float* __restrict__ X,      // (1024, 4096) -- only row 0 used
    const float* __restrict__ W,      // (4096, 4095) row-major
    float* __restrict__ theta)        // (4095,)
{
    __shared__ float xs[4096];
    __shared__ float part[8][16];

    const int tid = threadIdx.x;

    // cooperative load of x (16 KB) into LDS
    #pragma unroll
    for (int i = tid; i < 4096; i += 256) xs[i] = X[i];
    __syncthreads();

    const int widx = tid >> 5;        // wave in block: 0..7 -> K chunk
    const int lane = tid & 31;
    const int h    = lane >> 4;       // half-wave select (K offset 0 or 2)
    const int nn   = lane & 15;       // column within tile
    const int tile = blockIdx.x;      // 0..255
    const int j    = tile * 16 + nn;
    const int jl   = (j < 4095) ? j : 4094;   // tile 255 col 15 is padding

    const int k0 = widx * 512;
    // B layout (4x16, f32): VGPR0 -> rows K = 0+2h, VGPR1 -> rows K = 1+2h
    const float* wp = W + (size_t)(k0 + 2 * h) * 4095 + jl;
    const float* xp = xs + k0 + 2 * h;

    v8f c = {0.f, 0.f, 0.f, 0.f, 0.f, 0.f, 0.f, 0.f};

    #pragma unroll 4
    for (int it = 0; it < 128; ++it) {
        v2f a;
        a.x = xp[0];                  // A[m][k+2h]   (same for all rows m)
        a.y = xp[1];                  // A[m][k+2h+1]
        v2f b;
        b.x = wp[0];                  // B[k+2h  ][n]
        b.y = wp[4095];               // B[k+2h+1][n]
        // D = A(16x4) * B(4x16) + C, f32
        c = __builtin_amdgcn_wmma_f32_16x16x4_f32(
                false, a, false, b, (short)0, c, false, false);
        wp += (size_t)4 * 4095;
        xp += 4;
    }

    // Row 0 of C lives in accumulator VGPR0 on lanes 0..15.
    if (lane < 16) part[widx][lane] = c[0];
    __syncthreads();

    // reduce the 8 K-chunk partials for this tile
    if (tid < 16) {
        float s = 0.f;
        #pragma unroll
        for (int w = 0; w < 8; ++w) s += part[w][tid];
        const int jj = tile * 16 + tid;
        if (jj < 4095) theta[jj] = s;
    }
}

// ---------------------------------------------------------------------------
// Kernel 2: phase = S @ (0.5*sqrt(0.5)*theta) via fast Walsh-Hadamard,
// then psi[x] = exp(i*phase[x]) / 64, written as interleaved complex64.
// Gray-code mask for RZ gate j (0-based, n=12):
//   i  = floor(log2(j+1)) + 1          (block index)
//   p  = j - (2^(i-1) - 1)             (position in block)
//   g  = gray((p+1) mod 2^(i-1))
//   mask = (g << (n-i+1)) | (1 << (n-i))
// S[x,j] = 2*parity(x & mask_j) - 1 = -(-1)^popcount(x&mask_j)
//   => phase = -WHT(t), t[mask_j] = 0.5*sqrt(gamma)*dot_j
// ---------------------------------------------------------------------------
__global__ __launch_bounds__(1024) void qeff_wht_exp(
    const float* __restrict__ theta,  // (4095,)
    float* __restrict__ out)          // (4096,) complex64 -> 8192 f32
{
    __shared__ float t[4096];
    const int tid = threadIdx.x;

    #pragma unroll
    for (int q = 0; q < 4; ++q) t[tid + q * 1024] = 0.f;  // t[0] stays 0
    __syncthreads();

    const float scale = 0.35355339059327373f;  // 0.5 * sqrt(0.5)
    #pragma unroll
    for (int j = tid; j < 4095; j += 1024) {
        const int i  = 32 - __clz(j + 1);
        const int sz = 1 << (i - 1);
        const int p  = j - (sz - 1);
        const int v  = (p + 1) & (sz - 1);
        const int g  = v ^ (v >> 1);
        const int m  = (g << (12 - i + 1)) | (1 << (12 - i));
        t[m] = scale * theta[j];   // masks form a permutation of 1..4095
    }
    __syncthreads();

    // 12-stage in-place WHT (pairs are disjoint within a stage)
    for (int s = 0; s < 12; ++s) {
        const int stride = 1 << s;
        #pragma unroll
        for (int r = 0; r < 2; ++r) {
            const int idx  = tid + r * 1024;
            const int aIdx = ((idx >> s) << (s + 1)) | (idx & (stride - 1));
            const int bIdx = aIdx | stride;
            const float u = t[aIdx];
            const float w = t[bIdx];
            t[aIdx] = u + w;
            t[bIdx] = u - w;
        }
        __syncthreads();
    }

    const float inv = 0.015625f;  // 1/sqrt(4096)
    #pragma unroll
    for (int q = 0; q < 4; ++q) {
        const int x = tid + q * 1024;
        const float ph = -t[x];
        float sn, cs;
        __sincosf(ph, &sn, &cs);
        out[2 * x]     = cs * inv;
        out[2 * x + 1] = sn * inv;
    }
}

// ---------------------------------------------------------------------------
extern "C" void kernel_launch(void* const* d_in, const int* in_sizes, int n_in,
                              void* d_out, int out_size, void* d_ws, size_t ws_size,
                              hipStream_t stream) {
    const float* X = (const float*)d_in[0];   // inputs (1024, 4096) f32
    const float* W = (const float*)d_in[1];   // qeff_weights (4096, 4095) f32
    float* out   = (float*)d_out;             // (1, 4096) complex64 interleaved
    float* theta = (float*)d_ws;              // 4095 f32 scratch

    qeff_gemv_wmma<<<256, 256, 0, stream>>>(X, W, theta);
    qeff_wht_exp<<<1, 1024, 0, stream>>>(theta, out);
}